// AttentionModule_39548058861674
// MI455X (gfx1250) — compile-verified
//
#include <hip/hip_runtime.h>

// ---------------------------------------------------------------------------
// MI455X (gfx1250) implementation.
//
// Bandwidth analysis: the einsum over v dominates (257 MB read + 115 MB
// written, ~16us at 23.3 TB/s). Those kernels are pure b128 streams with
// non-temporal hints. The only real GEMM (the 10->128->16 MLP) is mapped onto
// V_WMMA_F32_16X16X4_F32 (f32 WMMA keeps reference precision; K=10 pads to
// 3 k-steps), 20 single-wave workgroups, 16 rows each.
//
// v2 change: K-padding now uses clamped indices + 0/1 float masks instead of
// predicated loads, removing all s_and_saveexec/exec-restore sequences from
// the WMMA region (straight-line wave code; EXEC stays all-ones).
// ---------------------------------------------------------------------------

typedef float v2f  __attribute__((ext_vector_type(2)));
typedef float v8f  __attribute__((ext_vector_type(8)));
typedef float f4v  __attribute__((ext_vector_type(4)));

#define DIMQ   10
#define VCH    5
#define PNUM   50176          // 224*224
#define CNUM   4
#define BNUM   64
#define H1     128
#define LN_EPS 1e-5f

// workspace layout (floats)
#define WS_QE   0             // 64  x 16
#define WS_KE   1024          // 256 x 16
#define WS_ATTN 5120          // 256
#define OUT_ELEMS (BNUM * VCH * PNUM)   // 16,056,320

// ---------------------------------------------------------------------------
// Kernel 1: MLP + LayerNorm embeddings via v_wmma_f32_16x16x4_f32.
// Grid: 20 blocks x 32 threads (1 wave). Blocks 0..3 -> q rows, 4..19 -> k.
// A-frag layout (ISA 7.12.2, 32-bit A 16x4): lanes 0-15 hold K=0,1 in
// (a.x,a.y); lanes 16-31 hold K=2,3. B-frag mirrors with N in the lane index
// (VGPR0 = row K=0 / K=2, VGPR1 = row K=1 / K=3, striped over lanes).
// C/D: VGPR i, lane l -> (M = i + 8*(l>=16), N = l&15).
// ---------------------------------------------------------------------------
__global__ __launch_bounds__(32) void mlp_embed_kernel(
    const float* __restrict__ q,  const float* __restrict__ k,
    const float* __restrict__ qw1, const float* __restrict__ qb1,
    const float* __restrict__ qw2, const float* __restrict__ qb2,
    const float* __restrict__ qg,  const float* __restrict__ qbe,
    const float* __restrict__ kw1, const float* __restrict__ kb1,
    const float* __restrict__ kw2, const float* __restrict__ kb2,
    const float* __restrict__ kg,  const float* __restrict__ kbe,
    float* __restrict__ ws)
{
    __shared__ float hs[16 * 132];   // 16x128 hidden, stride 132 kills bank conflicts
    __shared__ float s2[16 * 17];    // 16x16 layer-2 out, stride 17

    const int lane = threadIdx.x;
    const int half = lane >> 4;      // 0 | 1
    const int l15  = lane & 15;
    const int r0   = blockIdx.x * 16;
    const bool isQ = (r0 < BNUM);

    const float* w1 = isQ ? qw1 : kw1;
    const float* b1 = isQ ? qb1 : kb1;
    const float* w2 = isQ ? qw2 : kw2;
    const float* b2 = isQ ? qb2 : kb2;
    const float* g  = isQ ? qg  : kg;
    const float* be = isQ ? qbe : kbe;

    const int row = r0 + l15;        // global row 0..319

    // ---- A fragments for layer 1 (x rows, K = 10 padded to 12) ----
    // Branch-free padding: clamp index into range, zero via 0/1 mask.
    v2f a[3];
    #pragma unroll
    for (int kk = 0; kk < 3; ++kk) {
        const int k0  = kk * 4 + 2 * half;           // {0,2,4,6,8,10}
        const int kc0 = (k0     < DIMQ) ? k0     : 0;
        const int kc1 = (k0 + 1 < DIMQ) ? k0 + 1 : 0;
        const float m = (k0 < DIMQ) ? 1.f : 0.f;     // k0,k0+1 valid together
        float x0, x1;
        if (isQ) {                                    // uniform branch
            // q: (B,10,1) -> b*10 + d
            x0 = q[row * DIMQ + kc0];
            x1 = q[row * DIMQ + kc1];
        } else {
            // k: (B,10,1,C) -> b*40 + d*4 + c
            const int rr = row - BNUM;
            const int bb = rr >> 2, cc = rr & 3;
            x0 = k[bb * DIMQ * CNUM + kc0 * CNUM + cc];
            x1 = k[bb * DIMQ * CNUM + kc1 * CNUM + cc];
        }
        a[kk].x = x0 * m;
        a[kk].y = x1 * m;
    }

    // ---- layer 1: h = leaky_relu(x @ w1^T + b1), 8 tiles of N=16 ----
    #pragma unroll
    for (int nt = 0; nt < 8; ++nt) {
        const int n = nt * 16 + l15;
        v8f acc = {};
        #pragma unroll
        for (int kk = 0; kk < 3; ++kk) {
            const int k0  = kk * 4 + 2 * half;
            const int kc0 = (k0     < DIMQ) ? k0     : 0;
            const int kc1 = (k0 + 1 < DIMQ) ? k0 + 1 : 0;
            const float m = (k0 < DIMQ) ? 1.f : 0.f;
            v2f bf;
            bf.x = w1[n * DIMQ + kc0] * m;
            bf.y = w1[n * DIMQ + kc1] * m;
            acc = __builtin_amdgcn_wmma_f32_16x16x4_f32(
                false, a[kk], false, bf, (short)0, acc, false, false);
        }
        const float bias = b1[n];
        #pragma unroll
        for (int i = 0; i < 8; ++i) {
            float hv = acc[i] + bias;
            hv = (hv >= 0.f) ? hv : 0.1f * hv;          // LeakyReLU(0.1)
            const int m = i + 8 * half;
            hs[m * 132 + nt * 16 + l15] = hv;
        }
    }
    __syncthreads();

    // ---- layer 2: s = h @ w2^T + b2, K = 128 (32 wmma steps) ----
    v8f acc2 = {};
    #pragma unroll
    for (int kk = 0; kk < 32; ++kk) {
        const int k0 = kk * 4 + 2 * half;
        v2f af, bf;
        af.x = hs[l15 * 132 + k0];
        af.y = hs[l15 * 132 + k0 + 1];
        bf.x = w2[l15 * H1 + k0];        // w2 is (16,128); B = w2^T
        bf.y = w2[l15 * H1 + k0 + 1];
        acc2 = __builtin_amdgcn_wmma_f32_16x16x4_f32(
            false, af, false, bf, (short)0, acc2, false, false);
    }
    {
        const float bias2 = b2[l15];
        #pragma unroll
        for (int i = 0; i < 8; ++i) {
            const int m = i + 8 * half;
            s2[m * 17 + l15] = acc2[i] + bias2;
        }
    }
    __syncthreads();

    // ---- LayerNorm(16) per row; lanes 0..15 each own one row ----
    if (lane < 16) {
        const int m = lane;
        float vals[16];
        float mu = 0.f;
        #pragma unroll
        for (int n = 0; n < 16; ++n) { vals[n] = s2[m * 17 + n]; mu += vals[n]; }
        mu *= (1.f / 16.f);
        float var = 0.f;
        #pragma unroll
        for (int n = 0; n < 16; ++n) { const float d = vals[n] - mu; var += d * d; }
        var *= (1.f / 16.f);
        const float rs = rsqrtf(var + LN_EPS);
        const int gr = r0 + m;
        float* dst = (gr < BNUM) ? (ws + WS_QE + gr * 16)
                                 : (ws + WS_KE + (gr - BNUM) * 16);
        #pragma unroll
        for (int n = 0; n < 16; ++n)
            dst[n] = (vals[n] - mu) * rs * g[n] + be[n];
    }
}

// ---------------------------------------------------------------------------
// Kernel 2: logits + modality dropout + softmax(T=10) -> attn (B,C)
// ---------------------------------------------------------------------------
__global__ __launch_bounds__(64) void softmax_kernel(
    const float* __restrict__ md, float* __restrict__ ws)
{
    const int b = threadIdx.x;               // 0..63
    const float* qe = ws + WS_QE + b * 16;
    float logit[CNUM];
    #pragma unroll
    for (int c = 0; c < CNUM; ++c) {
        const float* ke = ws + WS_KE + (b * CNUM + c) * 16;
        float d = 0.f;
        #pragma unroll
        for (int n = 0; n < 16; ++n) d += qe[n] * ke[n];
        // scale = 10^-0.5 ; dropout mask ; /TEMPERATURE
        logit[c] = (d * 0.31622776601683794f - md[b * CNUM + c] * 1e5f) * 0.1f;
    }
    float mx = logit[0];
    #pragma unroll
    for (int c = 1; c < CNUM; ++c) mx = fmaxf(mx, logit[c]);
    float e[CNUM], s = 0.f;
    #pragma unroll
    for (int c = 0; c < CNUM; ++c) { e[c] = __expf(logit[c] - mx); s += e[c]; }
    const float inv = 1.f / s;
    #pragma unroll
    for (int c = 0; c < CNUM; ++c) ws[WS_ATTN + b * CNUM + c] = e[c] * inv;
}

// ---------------------------------------------------------------------------
// Kernel 3: out[b,v,p] = dot(attn[b,:], v[b,v,p,:])  -- the HBM stream.
// Grid (196, B*VCH) x 256: exact cover of P. One b128 NT load + dot + NT store.
// attn address is uniform per block -> scalar (s_load_b128) broadcast.
// ---------------------------------------------------------------------------
__global__ __launch_bounds__(256) void weighted_sum_kernel(
    const float* __restrict__ v, const float* __restrict__ ws,
    float* __restrict__ out)
{
    const int p  = blockIdx.x * 256 + threadIdx.x;   // < PNUM
    const int bv = blockIdx.y;                       // b*VCH + vch
    const int b  = bv / VCH;                         // uniform
    const f4v attn = *(const f4v*)(ws + WS_ATTN + b * CNUM);
    const size_t idx = (size_t)bv * PNUM + p;        // out element; v = idx*4
    const f4v vv = __builtin_nontemporal_load((const f4v*)v + idx);
    const float r = vv.x * attn.x + vv.y * attn.y + vv.z * attn.z + vv.w * attn.w;
    __builtin_nontemporal_store(r, out + idx);
}

// ---------------------------------------------------------------------------
// Kernel 4: attn_map[b,c,p] = attn[b,c]  (broadcast, float4 NT stores)
// Grid (49, B*C) x 256: exact cover of P/4.
// ---------------------------------------------------------------------------
__global__ __launch_bounds__(256) void attn_map_kernel(
    const float* __restrict__ ws, float* __restrict__ amap)
{
    const int p4 = blockIdx.x * 256 + threadIdx.x;   // < PNUM/4
    const int bc = blockIdx.y;                       // b*C + c
    const float a = ws[WS_ATTN + bc];                // uniform -> scalar load
    const f4v val = {a, a, a, a};
    __builtin_nontemporal_store(val, (f4v*)amap + (size_t)bc * (PNUM / 4) + p4);
}

// ---------------------------------------------------------------------------
extern "C" void kernel_launch(void* const* d_in, const int* in_sizes, int n_in,
                              void* d_out, int out_size, void* d_ws, size_t ws_size,
                              hipStream_t stream)
{
    const float* q    = (const float*)d_in[0];
    const float* k    = (const float*)d_in[1];
    const float* v    = (const float*)d_in[2];
    const float* md   = (const float*)d_in[3];
    const float* qw1  = (const float*)d_in[4];
    const float* qb1  = (const float*)d_in[5];
    const float* qw2  = (const float*)d_in[6];
    const float* qb2  = (const float*)d_in[7];
    const float* qg   = (const float*)d_in[8];
    const float* qbe  = (const float*)d_in[9];
    const float* kw1  = (const float*)d_in[10];
    const float* kb1  = (const float*)d_in[11];
    const float* kw2  = (const float*)d_in[12];
    const float* kb2  = (const float*)d_in[13];
    const float* kg   = (const float*)d_in[14];
    const float* kbe  = (const float*)d_in[15];

    float* ws   = (float*)d_ws;
    float* out  = (float*)d_out;                 // (B,VCH,224,224)
    float* amap = out + OUT_ELEMS;               // (B,C,224,224)

    mlp_embed_kernel<<<20, 32, 0, stream>>>(
        q, k, qw1, qb1, qw2, qb2, qg, qbe,
        kw1, kb1, kw2, kb2, kg, kbe, ws);

    softmax_kernel<<<1, 64, 0, stream>>>(md, ws);

    weighted_sum_kernel<<<dim3(PNUM / 256, BNUM * VCH), 256, 0, stream>>>(v, ws, out);

    attn_map_kernel<<<dim3(PNUM / (4 * 256), BNUM * CNUM), 256, 0, stream>>>(ws, amap);
}